// BiologicalWorkingMemory_87935160418554
// MI455X (gfx1250) — compile-verified
//
#include <hip/hip_runtime.h>

typedef float v2f __attribute__((ext_vector_type(2)));
typedef float v8f __attribute__((ext_vector_type(8)));

static constexpr int  BATCH = 65536;
static constexpr int  NS    = 8;
static constexpr int  NDIM  = 64;
static constexpr long long OUT_A   = (long long)BATCH * NS * NDIM;     // a
static constexpr long long OUT_G   = OUT_A  + (long long)BATCH * NS;   // g
static constexpr long long OUT_MC  = OUT_G  + (long long)BATCH * NS;   // mc
static constexpr long long OUT_LD  = OUT_MC + (long long)BATCH * NS;   // memory_load
static constexpr long long OUT_TOT = OUT_LD  + BATCH;                  // total_activity
static constexpr long long OUT_MNT = OUT_TOT + BATCH;                  // maintenance_strength

// triangular index into symmetric 8x8 Gram (i<=j storage), 36 entries
__device__ __forceinline__ constexpr int TIX(int i, int j) {
  const int a = i < j ? i : j;
  const int b = i < j ? j : i;
  return a * 8 + b - (a * (a + 1)) / 2;
}

__device__ __forceinline__ float clamp01(float x) {
  return fminf(fmaxf(x, 0.0f), 1.0f);
}

__global__ __launch_bounds__(32) void bwm_kernel(
    const float* __restrict__ mem_slots,    // (B,S,D)
    const float* __restrict__ slot_act,     // (B,S)
    const float* __restrict__ slot_gates,   // (B,S)
    const float* __restrict__ gate_thr,     // (B,S)
    const float* __restrict__ refresh_str,  // (B,S)
    const float* __restrict__ interf,       // (B,S,S)
    const float* __restrict__ maint_cur,    // (B,S)
    const float* __restrict__ inputs,       // (B,D)
    const float* __restrict__ gate_sig,     // (B,S)
    const float* __restrict__ refresh_sig,  // (B,S)
    const int*   __restrict__ dt_ptr,       // scalar
    float* __restrict__ out)
{
  // one wave32 block handles 4 batches: lanes 8*hb..8*hb+7 -> batch hb
  __shared__ float smd[32][68];   // decayed M, 32 rows (4 batches x 8 slots), padded
  __shared__ float sW[4][68];     // interference matrices, pre-scaled by 0.1
  __shared__ float sG[32][17];    // WMMA Gram results (2 tiles), padded

  const int lane = (int)(threadIdx.x & 31u);
  const int hb   = lane >> 3;     // batch sub-index 0..3
  const int p    = lane & 7;      // owns dims 8p..8p+7
  const int hh   = lane >> 4;     // WMMA lane-half
  const int q2   = lane & 15;     // WMMA row-in-tile
  const int bb   = (int)blockIdx.x * 4 + hb;

  // ---- load + decay memory rows; stage into LDS ----
  float4 m[NS][2];
  const float4* mg = reinterpret_cast<const float4*>(mem_slots);
#pragma unroll
  for (int s = 0; s < NS; ++s) {
    float4 v0 = mg[bb * 128 + s * 16 + 2 * p];
    float4 v1 = mg[bb * 128 + s * 16 + 2 * p + 1];
    v0.x *= 0.95f; v0.y *= 0.95f; v0.z *= 0.95f; v0.w *= 0.95f;
    v1.x *= 0.95f; v1.y *= 0.95f; v1.z *= 0.95f; v1.w *= 0.95f;
    m[s][0] = v0; m[s][1] = v1;
    *reinterpret_cast<float4*>(&smd[hb * 8 + s][8 * p])     = v0;
    *reinterpret_cast<float4*>(&smd[hb * 8 + s][8 * p + 4]) = v1;
  }
  { // stage per-batch interference matrix, pre-scaled by 0.1
    float4 w0 = reinterpret_cast<const float4*>(interf)[bb * 16 + 2 * p];
    float4 w1 = reinterpret_cast<const float4*>(interf)[bb * 16 + 2 * p + 1];
    w0.x *= 0.1f; w0.y *= 0.1f; w0.z *= 0.1f; w0.w *= 0.1f;
    w1.x *= 0.1f; w1.y *= 0.1f; w1.z *= 0.1f; w1.w *= 0.1f;
    *reinterpret_cast<float4*>(&sW[hb][8 * p])     = w0;
    *reinterpret_cast<float4*>(&sW[hb][8 * p + 4]) = w1;
  }

  // ---- per-slot scalars (redundant per lane within batch group) ----
  float a0[NS], thr[NS], rst[NS], mcu[NS], rsg[NS], g[NS];
#define LOAD8(dst, src) { const float4* p_ = reinterpret_cast<const float4*>(src); \
  float4 lo_ = p_[bb * 2], hi_ = p_[bb * 2 + 1];                                    \
  dst[0]=lo_.x; dst[1]=lo_.y; dst[2]=lo_.z; dst[3]=lo_.w;                           \
  dst[4]=hi_.x; dst[5]=hi_.y; dst[6]=hi_.z; dst[7]=hi_.w; }
  LOAD8(a0,  slot_act)
  LOAD8(thr, gate_thr)
  LOAD8(rst, refresh_str)
  LOAD8(mcu, maint_cur)
  LOAD8(rsg, refresh_sig)
  { // fold gates + gate_signals into g immediately (frees both operands)
    float gat[NS], gsg[NS];
    LOAD8(gat, slot_gates)
    LOAD8(gsg, gate_sig)
#pragma unroll
    for (int s = 0; s < NS; ++s) g[s] = 0.7f * gat[s] + 0.3f * clamp01(gsg[s]);
  }
#undef LOAD8
  const float4 in4a = reinterpret_cast<const float4*>(inputs)[bb * 16 + 2 * p];
  const float4 in4b = reinterpret_cast<const float4*>(inputs)[bb * 16 + 2 * p + 1];

  __syncthreads();

  // ---- Grams of decayed rows via V_WMMA_F32_16X16X4_F32, two 16-row tiles ----
  // A layout (16x4 f32): lane L: M=L%16, vgpr0:K=2*(L>=16), vgpr1:K+1.
  // For a Gram, the 4x16 B layout maps to the same registers.
  v8f acc0 = {};
  v8f acc1 = {};
#pragma unroll
  for (int kc = 0; kc < 16; ++kc) {
    const int k0 = kc * 4 + hh * 2;
    v2f ab0, ab1;
    ab0.x = smd[q2][k0];      ab0.y = smd[q2][k0 + 1];
    ab1.x = smd[16 + q2][k0]; ab1.y = smd[16 + q2][k0 + 1];
    acc0 = __builtin_amdgcn_wmma_f32_16x16x4_f32(false, ab0, false, ab0,
                                                 (short)0, acc0, false, false);
    acc1 = __builtin_amdgcn_wmma_f32_16x16x4_f32(false, ab1, false, ab1,
                                                 (short)0, acc1, false, false);
  }
  // scatter C tiles (vgpr v: M = v + 8*(lane>=16), N = lane%16)
#pragma unroll
  for (int v = 0; v < 8; ++v) {
    sG[v + 8 * hh][q2]      = acc0[v];
    sG[16 + v + 8 * hh][q2] = acc1[v];
  }
  __syncthreads();

  // gather this batch's symmetric 8x8 Gram (broadcast LDS reads)
  float G[36];
#pragma unroll
  for (int i = 0; i < 8; ++i)
#pragma unroll
    for (int j = i; j < 8; ++j)
      G[TIX(i, j)] = sG[hb * 8 + i][(hb & 1) * 8 + j];

  // ---- decayed activities ----
  float a[NS]; bool act[NS];
#pragma unroll
  for (int s = 0; s < NS; ++s) { a[s] = a0[s] * 0.9f; act[s] = a[s] > 0.1f; }

  // ---- sequential interference loop with incremental Gram maintenance ----
#pragma unroll
  for (int i = 0; i < 8; ++i) {
#pragma unroll
    for (int j = 0; j < 8; ++j) {
      if (i == j) continue;
      const float gij = G[TIX(i, j)];
      const float gii = G[TIX(i, i)];
      const float gjj = G[TIX(j, j)];
      const float den = __builtin_amdgcn_sqrtf(gii * gjj) + 1e-6f;
      const float sim = gij * __builtin_amdgcn_rcpf(den);
      const float c   = (act[i] && act[j]) ? sW[hb][i * 8 + j] * sim : 0.0f;
      // m_i -= c * m_j   (this lane's 8 dims, packs into v_pk_fma_f32)
      m[i][0].x -= c * m[j][0].x; m[i][0].y -= c * m[j][0].y;
      m[i][0].z -= c * m[j][0].z; m[i][0].w -= c * m[j][0].w;
      m[i][1].x -= c * m[j][1].x; m[i][1].y -= c * m[j][1].y;
      m[i][1].z -= c * m[j][1].z; m[i][1].w -= c * m[j][1].w;
      // G row/col i update (symmetric storage), exact rank-1 algebra
#pragma unroll
      for (int k = 0; k < 8; ++k)
        if (k != i) G[TIX(i, k)] -= c * G[TIX(j, k)];
      G[TIX(i, i)] = fmaf(c, fmaf(c, gjj, -2.0f * gij), gii);
    }
  }

  // ---- target slot selection (argmin, ties -> lowest index) ----
  bool anyav = false;
  float bm = __builtin_inff(); int tsm = 0;
  float bu = __builtin_inff(); int tsu = 0;
#pragma unroll
  for (int s = 0; s < 8; ++s) {
    const bool av = a0[s] < 0.2f;
    anyav = anyav || av;
    const float mv = av ? a0[s] : __builtin_inff();
    if (mv < bm)    { bm = mv;    tsm = s; }
    if (a0[s] < bu) { bu = a0[s]; tsu = s; }
  }
  const int ts = anyav ? tsm : tsu;

  // ---- write decision ----
  float gsv = g[0], thv = thr[0];
#pragma unroll
  for (int s = 1; s < NS; ++s) if (ts == s) { gsv = g[s]; thv = thr[s]; }
  const bool wr = gsv > thv;

  // ||inputs|| : 8-lane reduction within the batch group
  float n2 = in4a.x * in4a.x + in4a.y * in4a.y + in4a.z * in4a.z + in4a.w * in4a.w
           + in4b.x * in4b.x + in4b.y * in4b.y + in4b.z * in4b.z + in4b.w * in4b.w;
  n2 += __shfl_xor(n2, 1);
  n2 += __shfl_xor(n2, 2);
  n2 += __shfl_xor(n2, 4);
  const float innorm = __builtin_amdgcn_sqrtf(n2);

  const float bl   = gsv * 0.3f;
  const float fmix = 1.0f - bl;

  // ---- activity pipeline: write -> refresh -> mc -> capacity deactivation ----
  float rs[NS];
#pragma unroll
  for (int s = 0; s < NS; ++s) {
    if (wr && (ts == s)) a[s] = innorm;
    const float ru = clamp01(rsg[s]);
    rs[s] = (ru > 0.1f) ? rst[s] * ru : 0.0f;
    a[s] += rs[s];
  }

  const float dtf = (float)(*dt_ptr);
  float mc[NS];
#pragma unroll
  for (int s = 0; s < NS; ++s) {
    mc[s] = (a[s] > 0.1f) ? (mcu[s] + (a[s] * 0.5f - mcu[s]) * 0.1f * dtf)
                          : (mcu[s] * 0.95f);
  }

  int cnt = 0; bool actv[NS]; float mk[NS];
#pragma unroll
  for (int s = 0; s < NS; ++s) {
    actv[s] = a[s] > 0.1f;
    cnt += actv[s] ? 1 : 0;
    mk[s] = actv[s] ? a[s] : __builtin_inff();
  }
  const int nd = (cnt > 4) ? (cnt - 4) : 0;
  bool de[NS];
#pragma unroll
  for (int s = 0; s < NS; ++s) {
    int rank = 0;
#pragma unroll
    for (int k = 0; k < NS; ++k) {
      if (k == s) continue;
      rank += ((mk[k] < mk[s]) || (mk[k] == mk[s] && k < s)) ? 1 : 0;
    }
    de[s] = actv[s] && (rank < nd);
    if (de[s]) a[s] *= 0.5f;
  }

  // ---- fold blend / refresh-scale / deact-scale into one affine pass on m ----
  float4* outm = reinterpret_cast<float4*>(out);
#pragma unroll
  for (int s = 0; s < NS; ++s) {
    const bool wsel = wr && (ts == s);
    const float sc = (1.0f + rs[s]) * (de[s] ? 0.7f : 1.0f);
    const float al = (wsel ? fmix : 1.0f) * sc;
    const float be = (wsel ? bl : 0.0f) * sc;
    float4 o0, o1;
    o0.x = al * m[s][0].x + be * in4a.x;
    o0.y = al * m[s][0].y + be * in4a.y;
    o0.z = al * m[s][0].z + be * in4a.z;
    o0.w = al * m[s][0].w + be * in4a.w;
    o1.x = al * m[s][1].x + be * in4b.x;
    o1.y = al * m[s][1].y + be * in4b.y;
    o1.z = al * m[s][1].z + be * in4b.z;
    o1.w = al * m[s][1].w + be * in4b.w;
    outm[bb * 128 + s * 16 + 2 * p]     = o0;
    outm[bb * 128 + s * 16 + 2 * p + 1] = o1;
  }

  // ---- scalar outputs ----
  float suma = 0.0f, summc = 0.0f; int ldn = 0;
#pragma unroll
  for (int s = 0; s < NS; ++s) {
    suma += a[s]; summc += mc[s];
    ldn += (a[s] > 0.1f) ? 1 : 0;
  }

  if (p == 0) {  // one lane per batch stores the per-slot vectors
    float4* oa = reinterpret_cast<float4*>(out + OUT_A);
    oa[bb * 2]     = make_float4(a[0], a[1], a[2], a[3]);
    oa[bb * 2 + 1] = make_float4(a[4], a[5], a[6], a[7]);
    float4* og = reinterpret_cast<float4*>(out + OUT_G);
    og[bb * 2]     = make_float4(g[0], g[1], g[2], g[3]);
    og[bb * 2 + 1] = make_float4(g[4], g[5], g[6], g[7]);
    float4* oc = reinterpret_cast<float4*>(out + OUT_MC);
    oc[bb * 2]     = make_float4(mc[0], mc[1], mc[2], mc[3]);
    oc[bb * 2 + 1] = make_float4(mc[4], mc[5], mc[6], mc[7]);
    out[OUT_LD  + bb] = (float)ldn;
    out[OUT_TOT + bb] = suma;
    out[OUT_MNT + bb] = summc * 0.125f;
  }
}

extern "C" void kernel_launch(void* const* d_in, const int* in_sizes, int n_in,
                              void* d_out, int out_size, void* d_ws, size_t ws_size,
                              hipStream_t stream) {
  (void)in_sizes; (void)n_in; (void)out_size; (void)d_ws; (void)ws_size;
  const float* mem  = (const float*)d_in[0];
  const float* sact = (const float*)d_in[1];
  const float* sgat = (const float*)d_in[2];
  const float* gthr = (const float*)d_in[3];
  const float* rstr = (const float*)d_in[4];
  const float* intf = (const float*)d_in[5];
  const float* mcur = (const float*)d_in[6];
  const float* xin  = (const float*)d_in[7];
  const float* gsig = (const float*)d_in[8];
  const float* rsig = (const float*)d_in[9];
  const int*   dtp  = (const int*)  d_in[10];
  float* outp = (float*)d_out;

  dim3 grid(BATCH / 4);
  dim3 block(32);
  hipLaunchKernelGGL(bwm_kernel, grid, block, 0, stream,
                     mem, sact, sgat, gthr, rstr, intf, mcur, xin, gsig, rsig,
                     dtp, outp);
}